// GNN_61529701482954
// MI455X (gfx1250) — compile-verified
//
#include <hip/hip_runtime.h>

typedef __attribute__((ext_vector_type(2))) float v2f;
typedef __attribute__((ext_vector_type(8))) float v8f;

#define NN 100000
#define NE 1600000
#define IND 128
#define HID 128
#define OUTD 64
#define NG 512

// ---------------- degree / scatter kernels ----------------

__global__ void fill_zero_k(float* __restrict__ p, int n) {
    int i = blockIdx.x * blockDim.x + threadIdx.x;
    if (i < n) p[i] = 0.0f;
}

__global__ void deg_accum_k(const int* __restrict__ dst, float* __restrict__ deg, int E) {
    int e = blockIdx.x * blockDim.x + threadIdx.x;
    if (e < E) atomicAdd(&deg[dst[e]], 1.0f);
}

__global__ void deg_to_dinv_k(float* __restrict__ d, int n) {
    int i = blockIdx.x * blockDim.x + threadIdx.x;
    if (i < n) d[i] = rsqrtf(d[i] + 1.0f);  // +1 = self loop; deg >= 1 always
}

// one wave per edge; each lane handles 4 contiguous features (float4 gather + 4 f32 atomics)
__global__ void edge_scatter_k(const int* __restrict__ src, const int* __restrict__ dst,
                               const float* __restrict__ dinv, const float* __restrict__ h,
                               float* __restrict__ agg, int E) {
    unsigned t = blockIdx.x * blockDim.x + threadIdx.x;
    unsigned e = t >> 5;
    if (e >= (unsigned)E) return;
    unsigned q = (t & 31u) * 4u;
    int s = src[e], d = dst[e];
    float w = dinv[s] * dinv[d];
    float4 hv = *reinterpret_cast<const float4*>(h + (size_t)s * HID + q);
    float* ap = agg + (size_t)d * HID + q;
    atomicAdd(ap + 0, hv.x * w);
    atomicAdd(ap + 1, hv.y * w);
    atomicAdd(ap + 2, hv.z * w);
    atomicAdd(ap + 3, hv.w * w);
}

// global_add_pool with fused bias+ReLU: pooled[batch[i]][f] += relu(h[i][f] + b[f])
__global__ void pool_scatter_k(const int* __restrict__ batch, const float* __restrict__ h,
                               const float* __restrict__ b, float* __restrict__ pooled,
                               int nNodes) {
    unsigned t = blockIdx.x * blockDim.x + threadIdx.x;
    unsigned node = t >> 5;
    if (node >= (unsigned)nNodes) return;
    unsigned q = (t & 31u) * 4u;
    int g = batch[node];
    float4 hv = *reinterpret_cast<const float4*>(h + (size_t)node * HID + q);
    float4 bb = *reinterpret_cast<const float4*>(b + q);
    float* pp = pooled + (size_t)g * HID + q;
    atomicAdd(pp + 0, fmaxf(hv.x + bb.x, 0.0f));
    atomicAdd(pp + 1, fmaxf(hv.y + bb.y, 0.0f));
    atomicAdd(pp + 2, fmaxf(hv.z + bb.z, 0.0f));
    atomicAdd(pp + 3, fmaxf(hv.w + bb.w, 0.0f));
}

// ---------------- fused WMMA f32 GEMM ----------------
// C[M,N] = act(A)[M,K] @ B[K,N] (+biasOut), optionally agg = C * dinv^2 (self-loop init).
// grid.x = M/(16*MT); blockDim.x = 32*(N/16): wave w owns n-tile w, MT m-tiles per wave
// (B fragment loaded once per k-step, reused for MT WMMAs).
// N, K are compile-time so all fragment addresses fold to base + immediate offset
// (global_load with IOFFSET, no per-iteration 64-bit address math, no remainder loop).
// Full-precision f32 path: V_WMMA_F32_16X16X4_F32, K stepped by 4.
// Per-lane fragment layout (ISA 7.12.2, 32-bit A 16x4 / B 4x16 / C 16x16):
//   half = lane>>4, l16 = lane&15
//   A vgpr j: A[row16 + l16][k + 2*half + j]  -> contiguous float2 per lane
//   B vgpr j: B[k + 2*half + j][nBase + l16]
//   C vgpr v: C[row16 + 8*half + v][nBase + l16]
// ACT_A fuses the previous layer's bias+ReLU into this GEMM's A loads.
// SELFLOOP epilogue also writes agg[r][c] = val * dinv[r]^2; __syncthreads() guards the
// case agg aliases A (each block reads/writes only its own rows).
template <int MT, int N, int K, bool ACT_A, bool SELFLOOP, bool BIAS_OUT>
__global__ void gemm_k(const float* __restrict__ A, const float* __restrict__ B,
                       float* __restrict__ C, const float* __restrict__ actBias,
                       const float* __restrict__ dinv, float* __restrict__ agg,
                       const float* __restrict__ biasOut) {
    const int lane   = threadIdx.x & 31;
    const int wave   = threadIdx.x >> 5;
    const int half   = lane >> 4;
    const int l16    = lane & 15;
    const int nBase  = wave * 16;
    const int mTile0 = blockIdx.x * MT;

    const float* aBase = A + (size_t)(mTile0 * 16 + l16) * K + half * 2;
    const float* bBase = B + (size_t)(half * 2) * N + nBase + l16;
    const float* abBase = actBias + half * 2;  // dereferenced only when ACT_A

    v8f acc[MT];
#pragma unroll
    for (int t = 0; t < MT; ++t) {
        v8f z = {};
        acc[t] = z;
    }

#pragma unroll 4
    for (int k = 0; k < K; k += 4) {
        v2f b;
        b.x = bBase[k * N];
        b.y = bBase[k * N + N];
        float abx = 0.0f, aby = 0.0f;
        if (ACT_A) {
            abx = abBase[k];
            aby = abBase[k + 1];
        }
#pragma unroll
        for (int t = 0; t < MT; ++t) {
            v2f a;
            a.x = aBase[t * 16 * K + k];
            a.y = aBase[t * 16 * K + k + 1];
            if (ACT_A) {
                a.x = fmaxf(a.x + abx, 0.0f);
                a.y = fmaxf(a.y + aby, 0.0f);
            }
            acc[t] = __builtin_amdgcn_wmma_f32_16x16x4_f32(false, a, false, b, (short)0,
                                                           acc[t], false, false);
        }
    }

    if (SELFLOOP) __syncthreads();  // agg may alias A; all reads of this block's rows done

    float bv = BIAS_OUT ? biasOut[nBase + l16] : 0.0f;
#pragma unroll
    for (int t = 0; t < MT; ++t) {
        const int rBase = (mTile0 + t) * 16 + half * 8;
#pragma unroll
        for (int v = 0; v < 8; ++v) {
            const size_t idx = (size_t)(rBase + v) * N + nBase + l16;
            float val = acc[t][v] + bv;
            C[idx] = val;
            if (SELFLOOP) {
                float w = dinv[rBase + v];
                agg[idx] = val * w * w;
            }
        }
    }
}

// ---------------- driver ----------------

extern "C" void kernel_launch(void* const* d_in, const int* in_sizes, int n_in,
                              void* d_out, int out_size, void* d_ws, size_t ws_size,
                              hipStream_t stream) {
    const float* x     = (const float*)d_in[0];
    const int*   ei    = (const int*)d_in[1];
    const int*   batch = (const int*)d_in[2];
    const float* W1    = (const float*)d_in[3];
    const float* b1    = (const float*)d_in[4];
    const float* W2    = (const float*)d_in[5];
    const float* b2    = (const float*)d_in[6];
    const float* Wl    = (const float*)d_in[7];
    const float* bl    = (const float*)d_in[8];
    float* out = (float*)d_out;

    const int* src = ei;       // edge_index[0]
    const int* dst = ei + NE;  // edge_index[1]

    // workspace layout (16B+ aligned): dinv | bufH (51.2MB) | bufAgg (51.2MB) | pooled
    char* ws = (char*)d_ws;
    const size_t FEAT_BYTES = (size_t)NN * HID * sizeof(float);  // 51,200,000
    float* dinv   = (float*)(ws);
    float* bufH   = (float*)(ws + 524288);
    float* bufAgg = (float*)(ws + 524288 + FEAT_BYTES);
    float* pooled = (float*)(ws + 524288 + 2 * FEAT_BYTES);

    const int TPB = 256;

    // degree -> dinv
    fill_zero_k<<<(NN + TPB - 1) / TPB, TPB, 0, stream>>>(dinv, NN);
    deg_accum_k<<<(NE + TPB - 1) / TPB, TPB, 0, stream>>>(dst, dinv, NE);
    deg_to_dinv_k<<<(NN + TPB - 1) / TPB, TPB, 0, stream>>>(dinv, NN);

    // ---- layer 1: bufH = x@W1 ; bufAgg = bufH * dinv^2 (self-loop) ; scatter edges ----
    // NN/16 = 6250 m-tiles = 5 * 1250
    gemm_k<5, HID, IND, false, true, false><<<1250, 32 * (HID / 16), 0, stream>>>(
        x, W1, bufH, nullptr, dinv, bufAgg, nullptr);
    edge_scatter_k<<<(NE * 32 + TPB - 1) / TPB, TPB, 0, stream>>>(src, dst, dinv, bufH,
                                                                  bufAgg, NE);
    // bufAgg now holds pre-activation layer-1 output (bias/ReLU fused into GEMM2 loads)

    // ---- layer 2: bufH = relu(bufAgg + b1)@W2 ; bufAgg = bufH * dinv^2 ; scatter ----
    gemm_k<5, HID, HID, true, true, false><<<1250, 32 * (HID / 16), 0, stream>>>(
        bufAgg, W2, bufH, b1, dinv, bufAgg, nullptr);
    edge_scatter_k<<<(NE * 32 + TPB - 1) / TPB, TPB, 0, stream>>>(src, dst, dinv, bufH,
                                                                  bufAgg, NE);
    // bufAgg now holds pre-activation layer-2 output

    // ---- global add pool with fused bias+ReLU ----
    fill_zero_k<<<(NG * HID + TPB - 1) / TPB, TPB, 0, stream>>>(pooled, NG * HID);
    pool_scatter_k<<<(NN * 32 + TPB - 1) / TPB, TPB, 0, stream>>>(batch, bufAgg, b2,
                                                                  pooled, NN);

    // ---- head: out = pooled @ Wl + bl ----  (NG/16 = 32 m-tiles = 2 * 16)
    gemm_k<2, OUTD, HID, false, false, true><<<16, 32 * (OUTD / 16), 0, stream>>>(
        pooled, Wl, out, nullptr, nullptr, nullptr, bl);
}